// GraphLearningModule_MLP_32512902431529
// MI455X (gfx1250) — compile-verified
//
#include <hip/hip_runtime.h>
#include <math.h>

typedef __attribute__((ext_vector_type(2))) float v2f;
typedef __attribute__((ext_vector_type(8))) float v8f;

// ---------------------------------------------------------------------------
// Kernel 1: S[m][0:64]  = x[m] @ W1[:, :64]^T + b1   (s_i half, bias folded)
//           S[m][64:128]= x[m] @ W1[:, 64:]^T        (s_j half)
// m = b*1024 + node, A = x viewed as [2048 x 64], one 16x16 tile per wave,
// K=64 chained through 16x V_WMMA_F32_16X16X4_F32 (exact fp32 math).
// ---------------------------------------------------------------------------
__global__ void glm_gemm_s_kernel(const float* __restrict__ x,   // [2048][64]
                                  const float* __restrict__ W1,  // [64][128]
                                  const float* __restrict__ b1,  // [64]
                                  float* __restrict__ S)         // [2048][128]
{
    const int wave = threadIdx.x >> 5;            // 8 waves per block
    const int lane = threadIdx.x & 31;
    const int tile = blockIdx.x * 8 + wave;       // 128*8 = 1024 tiles
    const int m0 = (tile >> 3) * 16;              // row tile (2048/16 = 128)
    const int n0 = (tile & 7) * 16;               // col tile (128/16 = 8)

    const int mrow = m0 + (lane & 15);            // A: lanes 0-15 hold M=0..15
    const int half = lane >> 4;                   // upper half-wave: K pair +2
    const int ncol = n0 + (lane & 15);            // B/D column per lane

    // B matrix column ncol over t:  n<64 -> W1[n][t],  n>=64 -> W1[n-64][64+t]
    const float* wcol = (ncol < 64) ? (W1 + ncol * 128)
                                    : (W1 + (ncol - 64) * 128 + 64);

    v8f acc = {};
    #pragma unroll
    for (int k0 = 0; k0 < 64; k0 += 4) {
        const int ka = k0 + half * 2;
        v2f a, b;
        a.x = x[mrow * 64 + ka];        // A 16x4 f32: VGPR0 K={0|2}, VGPR1 K={1|3}
        a.y = x[mrow * 64 + ka + 1];
        b.x = wcol[ka];                 // B 4x16 f32: row K striped across lanes
        b.y = wcol[ka + 1];
        acc = __builtin_amdgcn_wmma_f32_16x16x4_f32(
            /*neg_a=*/false, a, /*neg_b=*/false, b,
            /*c_mod=*/(short)0, acc, /*reuse_a=*/false, /*reuse_b=*/false);
    }

    // D layout: VGPR v -> M = v + (half?8:0), N = lane&15
    const float bias = (ncol < 64) ? b1[ncol] : 0.0f;
    const int rbase = m0 + (half ? 8 : 0);
    #pragma unroll
    for (int v = 0; v < 8; ++v)
        S[(rbase + v) * 128 + ncol] = acc[v] + bias;
}

// ---------------------------------------------------------------------------
// Kernel 2: one block per row i. For all j: logit = 0.5*(dot_b0 + dot_b1) + b2
// with leaky(u_b[h] + sj_b[j][h]) inner terms, + gumbel(i,j); block-wide
// argmax (first-index tie-break), emit one-hot row. S stays L2-resident.
// ---------------------------------------------------------------------------
__global__ void glm_row_argmax_kernel(const float* __restrict__ S,   // [2048][128]
                                      const float* __restrict__ W2,  // [64]
                                      const float* __restrict__ b2,  // [1]
                                      const float* __restrict__ gu,  // [1024][1024]
                                      float* __restrict__ out)       // [1024][1024]
{
    __shared__ float u0[64], u1[64], w2s[64];
    __shared__ float redv[256];
    __shared__ int   redi[256];

    const int i = blockIdx.x;
    const int t = threadIdx.x;

    if (t < 64) {
        u0[t]  = S[i * 128 + t];             // s_i(b=0) + b1 (folded)
        u1[t]  = S[(1024 + i) * 128 + t];    // s_i(b=1) + b1
        w2s[t] = W2[t];
    }
    __syncthreads();

    const float bb2 = b2[0];
    float bestv = -INFINITY;
    int   besti = 0;

    #pragma unroll
    for (int jj = 0; jj < 4; ++jj) {
        const int j = t + jj * 256;
        const float4* sj0 = (const float4*)(S + j * 128 + 64);          // b=0 s_j
        const float4* sj1 = (const float4*)(S + (1024 + j) * 128 + 64); // b=1 s_j
        float acc0 = 0.0f, acc1 = 0.0f;
        #pragma unroll
        for (int h4 = 0; h4 < 16; ++h4) {
            const float4 a0 = sj0[h4];
            const float4 a1 = sj1[h4];
            const int h = h4 * 4;
            float z;
            z = u0[h+0] + a0.x; z = z > 0.f ? z : 0.01f * z; acc0 = fmaf(z, w2s[h+0], acc0);
            z = u0[h+1] + a0.y; z = z > 0.f ? z : 0.01f * z; acc0 = fmaf(z, w2s[h+1], acc0);
            z = u0[h+2] + a0.z; z = z > 0.f ? z : 0.01f * z; acc0 = fmaf(z, w2s[h+2], acc0);
            z = u0[h+3] + a0.w; z = z > 0.f ? z : 0.01f * z; acc0 = fmaf(z, w2s[h+3], acc0);
            z = u1[h+0] + a1.x; z = z > 0.f ? z : 0.01f * z; acc1 = fmaf(z, w2s[h+0], acc1);
            z = u1[h+1] + a1.y; z = z > 0.f ? z : 0.01f * z; acc1 = fmaf(z, w2s[h+1], acc1);
            z = u1[h+2] + a1.z; z = z > 0.f ? z : 0.01f * z; acc1 = fmaf(z, w2s[h+2], acc1);
            z = u1[h+3] + a1.w; z = z > 0.f ? z : 0.01f * z; acc1 = fmaf(z, w2s[h+3], acc1);
        }
        const float u = gu[i * 1024 + j];
        const float g = -logf(-logf(u + 1e-10f) + 1e-10f);
        const float cand = 0.5f * (acc0 + acc1) + bb2 + g;
        if (cand > bestv) { bestv = cand; besti = j; }   // j strictly increasing
    }

    redv[t] = bestv; redi[t] = besti;
    __syncthreads();
    for (int s = 128; s > 0; s >>= 1) {
        if (t < s) {
            const float ov = redv[t + s]; const int oi = redi[t + s];
            if (ov > redv[t] || (ov == redv[t] && oi < redi[t])) {
                redv[t] = ov; redi[t] = oi;              // first-index tie-break
            }
        }
        __syncthreads();
    }
    const int best = redi[0];

    #pragma unroll
    for (int jj = 0; jj < 4; ++jj) {
        const int j = t + jj * 256;
        out[i * 1024 + j] = (j == best) ? 1.0f : 0.0f;
    }
}

// ---------------------------------------------------------------------------
extern "C" void kernel_launch(void* const* d_in, const int* in_sizes, int n_in,
                              void* d_out, int out_size, void* d_ws, size_t ws_size,
                              hipStream_t stream) {
    const float* x  = (const float*)d_in[0];   // [2,1024,64]
    const float* W1 = (const float*)d_in[1];   // [64,128]
    const float* b1 = (const float*)d_in[2];   // [64]
    const float* W2 = (const float*)d_in[3];   // [1,64]
    const float* b2 = (const float*)d_in[4];   // [1]
    const float* gu = (const float*)d_in[5];   // [1024,1024]
    float* out = (float*)d_out;                // [1024,1024]
    float* S   = (float*)d_ws;                 // [2048][128] = 1 MB scratch

    glm_gemm_s_kernel<<<128, 256, 0, stream>>>(x, W1, b1, S);
    glm_row_argmax_kernel<<<1024, 256, 0, stream>>>(S, W2, b2, gu, out);
}